// SynesthesiaCentralOrchestrator_89644557402438
// MI455X (gfx1250) — compile-verified
//
#include <hip/hip_runtime.h>
#include <math.h>

// Quaternion Karcher-mean orchestrator for MI455X (gfx1250, wave32).
// Pure elementwise/transcendental workload: no matrix structure -> no WMMA.
// Compute-bound (~7:1 vs 23.3 TB/s memory floor), so this round trims the
// VALU count per iteration:
//   * sinf(acosf(w)) -> sqrtf(1-w^2)            (exact identity on [0,pi])
//   * IEEE divides   -> v_rcp_f32 + mul          (~1 ulp, contractive loop)
//   * libm sin/cos   -> __sinf/__cosf (v_sin/v_cos) in the exp map
// Memory path: b128 non-temporal streaming loads/stores (336 MB stream >
// 192 MB L2), LDS block reduction + global_atomic_add_f32 for the scalar.

typedef float v4f __attribute__((ext_vector_type(4)));

#ifndef INV_PI_F
#define INV_PI_F 0.31830988618379067154f
#endif

__device__ __forceinline__ float rcpf_fast(float x) {
    return __builtin_amdgcn_rcpf(x);   // v_rcp_f32, ~1 ulp
}

__global__ void karcher_zero_acc(float* __restrict__ acc) {
    if (threadIdx.x == 0 && blockIdx.x == 0) *acc = 0.0f;
}

__global__ __launch_bounds__(256) void karcher_main(
    const v4f* __restrict__ audio,
    const v4f* __restrict__ vision,
    const v4f* __restrict__ text,
    const v4f* __restrict__ genomic,
    v4f*  __restrict__ bary,      // (B*L) float4
    float* __restrict__ eta,      // (B*L)
    float* __restrict__ curv_acc, // single float accumulator (d_ws)
    int N)
{
    const int tid = threadIdx.x;
    const int i   = blockIdx.x * blockDim.x + tid;

    float c2 = 0.0f;

    if (i < N) {
        // ---- streaming b128 loads, non-temporal ----
        v4f p[4];
        p[0] = __builtin_nontemporal_load(&audio[i]);
        p[1] = __builtin_nontemporal_load(&vision[i]);
        p[2] = __builtin_nontemporal_load(&text[i]);
        p[3] = __builtin_nontemporal_load(&genomic[i]);

        // quaternion_normalize: q / (||q|| + 1e-12)
        #pragma unroll
        for (int m = 0; m < 4; ++m) {
            float nrm = sqrtf(p[m].x*p[m].x + p[m].y*p[m].y +
                              p[m].z*p[m].z + p[m].w*p[m].w);
            float inv = rcpf_fast(nrm + 1e-12f);
            p[m].x *= inv; p[m].y *= inv; p[m].z *= inv; p[m].w *= inv;
        }

        // mu0 = points[0]
        float mw = p[0].x, mx = p[0].y, my = p[0].z, mz = p[0].w;

        // Reference early-out needs a GLOBAL norm over 12.6M residuals to
        // drop below 1e-6 -- unreachable in 10 iterations for random data,
        // so the reference deterministically runs all 10 -> fixed loop.
        for (int it = 0; it < 10; ++it) {
            float tvx = 0.0f, tvy = 0.0f, tvz = 0.0f;
            #pragma unroll
            for (int m = 0; m < 4; ++m) {
                // hamilton(conj(mu), p[m])
                const float w1 = mw, x1 = -mx, y1 = -my, z1 = -mz;
                const float w2 = p[m].x, x2 = p[m].y, y2 = p[m].z, z2 = p[m].w;
                const float rw = w1*w2 - x1*x2 - y1*y2 - z1*z2;
                const float rx = w1*x2 + x1*w2 + y1*z2 - z1*y2;
                const float ry = w1*y2 - x1*z2 + y1*w2 + z1*x2;
                const float rz = w1*z2 + x1*y2 - y1*x2 + z1*w2;
                // quaternion_log: theta/sin(theta) scale of the vector part.
                // sin(acos(wc)) == sqrt(1 - wc^2) exactly on [0,pi]; the
                // clamp guarantees 1 - wc^2 >= ~2e-7 > 0.
                const float wc = fminf(fmaxf(rw, -1.0f + 1e-7f), 1.0f - 1e-7f);
                const float th = acosf(wc);
                const float s  = sqrtf(1.0f - wc*wc) + 1e-9f;
                const float f  = 0.25f * th * rcpf_fast(s);  // weight 1/4 folded
                tvx = fmaf(f, rx, tvx);
                tvy = fmaf(f, ry, tvy);
                tvz = fmaf(f, rz, tvz);
            }
            // quaternion_exp(mean_v); th2 in [0, pi] -> fast v_sin/v_cos ok
            const float th2 = sqrtf(tvx*tvx + tvy*tvy + tvz*tvz);
            const float inv = rcpf_fast(th2 + 1e-9f);
            const float dw  = __cosf(th2);
            const float sf  = __sinf(th2) * inv;
            const float dx = sf*tvx, dy = sf*tvy, dz = sf*tvz;
            // mu_new = normalize(hamilton(mu, delta))
            const float nw = mw*dw - mx*dx - my*dy - mz*dz;
            const float nx = mw*dx + mx*dw + my*dz - mz*dy;
            const float ny = mw*dy - mx*dz + my*dw + mz*dx;
            const float nz = mw*dz + mx*dy - my*dx + mz*dw;
            const float nn   = sqrtf(nw*nw + nx*nx + ny*ny + nz*nz);
            const float ninv = rcpf_fast(nn + 1e-12f);
            mw = nw*ninv; mx = nx*ninv; my = ny*ninv; mz = nz*ninv;
        }

        // ---- outputs: barycenter (NT b128) ----
        v4f outq; outq.x = mw; outq.y = mx; outq.z = my; outq.w = mz;
        __builtin_nontemporal_store(outq, &bary[i]);

        // ---- spectral shift eta (full-precision, once per point) ----
        const float det_re = (mw*mw + mz*mz) - (-(mx*mx) - my*my);
        const float det_im = (mw*(-mz) + mz*mw) - (mx*my + my*(-mx)); // exact 0
        const float e = atan2f(det_im, det_re) * INV_PI_F;
        __builtin_nontemporal_store(e, &eta[i]);

        // ---- curvature contribution (full-precision divide, once) ----
        const float qn = sqrtf(mw*mw + mx*mx + my*my + mz*mz);
        const float qi = 1.0f / (qn + 1e-12f);
        const float d0 = mw - mw*qi;
        const float d1 = mx - mx*qi;
        const float d2 = my - my*qi;
        const float d3 = mz - mz*qi;
        c2 = d0*d0 + d1*d1 + d2*d2 + d3*d3;
    }

    // ---- block reduction in LDS, one global f32 atomic per block ----
    __shared__ float red[256];
    red[tid] = c2;
    __syncthreads();
    #pragma unroll
    for (int s = 128; s > 0; s >>= 1) {
        if (tid < s) red[tid] += red[tid + s];
        __syncthreads();
    }
    if (tid == 0) {
#if defined(__HIP_DEVICE_COMPILE__)
        unsafeAtomicAdd(curv_acc, red[0]);   // -> global_atomic_add_f32
#else
        atomicAdd(curv_acc, red[0]);
#endif
    }
}

__global__ void karcher_finalize(const float* __restrict__ curv_acc,
                                 float* __restrict__ veracity) {
    if (threadIdx.x == 0 && blockIdx.x == 0) {
        const float curvature = sqrtf(*curv_acc);
        *veracity = 1.0f / (1.0f + curvature);
    }
}

extern "C" void kernel_launch(void* const* d_in, const int* in_sizes, int n_in,
                              void* d_out, int out_size, void* d_ws, size_t ws_size,
                              hipStream_t stream) {
    const int N = in_sizes[0] / 4;               // B*L points
    const v4f* audio   = (const v4f*)d_in[0];
    const v4f* vision  = (const v4f*)d_in[1];
    const v4f* text    = (const v4f*)d_in[2];
    const v4f* genomic = (const v4f*)d_in[3];

    float* out   = (float*)d_out;
    v4f*   bary  = (v4f*)out;                    // N*4 floats
    float* eta   = out + (size_t)N * 4;          // N floats
    float* ver   = out + (size_t)N * 5;          // 1 float
    float* acc   = (float*)d_ws;                 // scalar accumulator

    karcher_zero_acc<<<1, 1, 0, stream>>>(acc);

    const int threads = 256;                     // 8 wave32s per block
    const int blocks  = (N + threads - 1) / threads;
    karcher_main<<<blocks, threads, 0, stream>>>(audio, vision, text, genomic,
                                                 bary, eta, acc, N);

    karcher_finalize<<<1, 1, 0, stream>>>(acc, ver);
}